// RBF_68968584839825
// MI455X (gfx1250) — compile-verified
//
#include <hip/hip_runtime.h>
#include <math.h>
#include <stdint.h>

typedef float v2f __attribute__((ext_vector_type(2)));
typedef float v8f __attribute__((ext_vector_type(8)));

#define KDIM 128
#define ODIM 512
#define LDS_STRIDE 132   // 128 + 4 pad: 16B-aligned rows, breaks 64-bank stride conflicts

// CDNA5 async global->LDS DMA (ASYNCcnt-tracked, no VGPR data movement).
// GVS mode: 64-bit SGPR base + per-lane 32-bit VGPR byte offset; LDS dest is a
// per-lane byte offset VGPR (low 32 bits of the LDS aperture address).
__device__ __forceinline__ void async_copy_b128(uint32_t lds_off, uint32_t g_off,
                                                uint64_t base) {
    asm volatile("global_load_async_to_lds_b128 %0, %1, %2"
                 :: "v"(lds_off), "v"(g_off), "s"(base) : "memory");
}
__device__ __forceinline__ void wait_async_zero() {
    asm volatile("s_wait_asynccnt 0" ::: "memory");
}

// Block tile: 64 rows x 128 cols. 256 threads = 8 waves (wave32).
// Wave tile: 16 rows x 64 cols = 4 WMMA C tiles of 16x16, K=128 via 32 x V_WMMA_F32_16X16X4_F32.
__global__ __launch_bounds__(256) void rbf_wmma_kernel(
    const float* __restrict__ x,
    const float* __restrict__ centers,
    const float* __restrict__ log_sigmas,
    float* __restrict__ out)
{
    extern __shared__ float lds[];
    float* sA  = lds;                       // 64  * 132 floats (x tile, row-major [row][k])
    float* sC  = sA + 64 * LDS_STRIDE;      // 128 * 132 floats (centers tile, [col][k])
    float* sXQ = sC + 128 * LDS_STRIDE;     // 64  row norms
    float* sCQ = sXQ + 64;                  // 128 col norms

    const int tid     = threadIdx.x;
    const int row_blk = blockIdx.y * 64;
    const int col_blk = blockIdx.x * 128;

    // LDS byte offsets (addrspace(3) offset = low 32 bits of generic pointer)
    const uint32_t ldsBase  = (uint32_t)(uintptr_t)(void*)lds;
    const uint32_t sC_off   = ldsBase + 64u * LDS_STRIDE * 4u;
    const uint64_t gxBase   = (uint64_t)(uintptr_t)(x + (size_t)row_blk * KDIM);
    const uint64_t gcBase   = (uint64_t)(uintptr_t)(centers + (size_t)col_blk * KDIM);

    // ---- async DMA stage: x tile (64 rows x 128 k), 16B per lane per issue ----
    #pragma unroll
    for (int j = 0; j < 8; ++j) {
        int idx = tid + 256 * j;                  // 0..2047 float4s
        int r   = idx >> 5;                       // row 0..63
        int q   = idx & 31;                       // float4 within row
        async_copy_b128(ldsBase + (uint32_t)(r * (LDS_STRIDE * 4) + q * 16),
                        (uint32_t)(r * (KDIM * 4) + q * 16), gxBase);
    }
    // ---- async DMA stage: centers tile (128 cols x 128 k) ----
    #pragma unroll
    for (int j = 0; j < 16; ++j) {
        int idx = tid + 256 * j;                  // 0..4095 float4s
        int r   = idx >> 5;                       // col 0..127
        int q   = idx & 31;
        async_copy_b128(sC_off + (uint32_t)(r * (LDS_STRIDE * 4) + q * 16),
                        (uint32_t)(r * (KDIM * 4) + q * 16), gcBase);
    }
    wait_async_zero();      // our waves' async writes to LDS complete
    __syncthreads();        // visible to whole workgroup

    // ---- squared norms (||x_row||^2, ||c_col||^2) once per block ----
    if (tid < 64) {
        float s = 0.0f;
        #pragma unroll 8
        for (int k = 0; k < KDIM; ++k) { float v = sA[tid * LDS_STRIDE + k]; s = fmaf(v, v, s); }
        sXQ[tid] = s;
    } else if (tid < 192) {
        int c = tid - 64;
        float s = 0.0f;
        #pragma unroll 8
        for (int k = 0; k < KDIM; ++k) { float v = sC[c * LDS_STRIDE + k]; s = fmaf(v, v, s); }
        sCQ[c] = s;
    }
    __syncthreads();

    const int wave  = tid >> 5;
    const int lane  = tid & 31;
    const int wrow  = (wave >> 1) * 16;    // 0,16,32,48
    const int wcol  = (wave & 1) * 64;     // 0 or 64
    const int lhalf = lane >> 4;           // 0 | 1 (selects K offset for A/B, M+8 for C)
    const int lmod  = lane & 15;

    v8f acc0 = {}, acc1 = {}, acc2 = {}, acc3 = {};

    // A frag (16x4 f32): lanes 0-15 -> M=lane, K={0,1}; lanes 16-31 -> M=lane-16, K={2,3}
    const float* aBase = sA + (wrow + lmod) * LDS_STRIDE + 2 * lhalf;
    // B frag (4x16 f32): VGPR v -> K=v+2*lhalf, N=lmod  (centers stored [col][k])
    const float* bBase = sC + (wcol + lmod) * LDS_STRIDE + 2 * lhalf;

    #pragma unroll
    for (int k0 = 0; k0 < KDIM; k0 += 4) {
        v2f a  = *(const v2f*)(aBase + k0);
        v2f b0 = *(const v2f*)(bBase + 0 * 16 * LDS_STRIDE + k0);
        v2f b1 = *(const v2f*)(bBase + 1 * 16 * LDS_STRIDE + k0);
        v2f b2 = *(const v2f*)(bBase + 2 * 16 * LDS_STRIDE + k0);
        v2f b3 = *(const v2f*)(bBase + 3 * 16 * LDS_STRIDE + k0);
        acc0 = __builtin_amdgcn_wmma_f32_16x16x4_f32(false, a, false, b0, (short)0, acc0, false, false);
        acc1 = __builtin_amdgcn_wmma_f32_16x16x4_f32(false, a, false, b1, (short)0, acc1, false, false);
        acc2 = __builtin_amdgcn_wmma_f32_16x16x4_f32(false, a, false, b2, (short)0, acc2, false, false);
        acc3 = __builtin_amdgcn_wmma_f32_16x16x4_f32(false, a, false, b3, (short)0, acc3, false, false);
    }

    // ---- fused RBF epilogue + store (lanes 0-15 -> consecutive cols: coalesced) ----
    v8f accs[4] = {acc0, acc1, acc2, acc3};
    #pragma unroll
    for (int t = 0; t < 4; ++t) {
        int   col_l = wcol + t * 16 + lmod;
        int   col   = col_blk + col_l;
        float csq   = sCQ[col_l];
        float inv_s = 1.0f / expf(log_sigmas[col]);
        #pragma unroll
        for (int r = 0; r < 8; ++r) {
            int   row_l = wrow + r + 8 * lhalf;       // C layout: VGPR r -> M=r (+8 for lanes 16-31)
            float cross = accs[t][r];
            float d2    = fmaxf(sXQ[row_l] - 2.0f * cross + csq, 0.0f);
            float dist  = sqrtf(d2) * inv_s;
            out[(size_t)(row_blk + row_l) * ODIM + col] = expf(-dist * dist);
        }
    }
}

extern "C" void kernel_launch(void* const* d_in, const int* in_sizes, int n_in,
                              void* d_out, int out_size, void* d_ws, size_t ws_size,
                              hipStream_t stream) {
    const float* x          = (const float*)d_in[0];
    const float* centers    = (const float*)d_in[1];
    const float* log_sigmas = (const float*)d_in[2];
    float* out              = (float*)d_out;

    const int n_rows = in_sizes[0] / KDIM;            // 131072
    dim3 grid(ODIM / 128, n_rows / 64);               // (4, 2048)
    dim3 block(256);
    size_t shmem = (64 + 128) * LDS_STRIDE * sizeof(float) + (64 + 128) * sizeof(float);
    rbf_wmma_kernel<<<grid, block, shmem, stream>>>(x, centers, log_sigmas, out);
}